// SinkhornDistance_84250078478898
// MI455X (gfx1250) — compile-verified
//
#include <hip/hip_runtime.h>

// Sinkhorn (reference-faithful, analytically collapsed):
//   Xs = softmax_rows(x); Ys = softmax_rows(y)
//   C  = x2[i] + y2[j] - 2 * Xs @ Ys^T                (f32 WMMA GEMM, TDM-fed LDS)
//   Lc[j] = logsumexp_i(-C[i,j]/EPS)
//   pi[i,j] = exp(LOG_NU - Lc[j] - C[i,j]/EPS)        (u+v == EPS*(log_nu - Lc), always)
//   cost = sum(pi * C)
// Outputs (flat): [cost (1) | pi (N*N) | C (N*N)]

#define N       4096
#define NNE     ((size_t)N * (size_t)N)
#define INV_EPS 10.0f
#define LOG_NU  (-8.3177252f)   // log(1/4096 + 1e-8)
#define SEG     16

#define BM 128
#define BN 128
#define BK 32
#define LDP 36                  // padded LDS row stride (dwords) == TDM pad 4dw per 32dw
#define PANEL_F (BM * LDP)      // floats per panel (A or B)

typedef float v2f __attribute__((ext_vector_type(2)));
typedef float v4f __attribute__((ext_vector_type(4)));
typedef float v8f __attribute__((ext_vector_type(8)));
typedef unsigned int u32x4 __attribute__((ext_vector_type(4)));
typedef int i32x4 __attribute__((ext_vector_type(4)));
typedef int i32x8 __attribute__((ext_vector_type(8)));

#if __has_builtin(__builtin_amdgcn_tensor_load_to_lds) && \
    __has_builtin(__builtin_amdgcn_s_wait_tensorcnt)
#define USE_TDM 1
#else
#define USE_TDM 0
#endif

#if __has_builtin(__builtin_amdgcn_s_wait_tensorcnt)
#define WAIT_TENSOR0() __builtin_amdgcn_s_wait_tensorcnt(0)
#else
#define WAIT_TENSOR0() asm volatile("s_wait_tensorcnt 0x0" ::: "memory")
#endif

__device__ __forceinline__ float blockMax256(float v, float* sm) {
    const int t = threadIdx.x;
    sm[t] = v; __syncthreads();
    for (int off = 128; off > 0; off >>= 1) {
        if (t < off) sm[t] = fmaxf(sm[t], sm[t + off]);
        __syncthreads();
    }
    float r = sm[0]; __syncthreads();
    return r;
}

__device__ __forceinline__ float blockSum256(float v, float* sm) {
    const int t = threadIdx.x;
    sm[t] = v; __syncthreads();
    for (int off = 128; off > 0; off >>= 1) {
        if (t < off) sm[t] = sm[t] + sm[t + off];
        __syncthreads();
    }
    float r = sm[0]; __syncthreads();
    return r;
}

// Row softmax + sum-of-squares per row. One block (256 threads) per row.
__global__ void __launch_bounds__(256)
softmax_rows_kernel(const float* __restrict__ in, float* __restrict__ out,
                    float* __restrict__ sq) {
    __shared__ float sm[256];
    const int row = blockIdx.x, t = threadIdx.x;
    const float* rin = in + (size_t)row * N;
    float* rout = out + (size_t)row * N;

    float m = -__builtin_inff();
    for (int c = t; c < N; c += 256) m = fmaxf(m, rin[c]);
    m = blockMax256(m, sm);

    float s = 0.f;
    for (int c = t; c < N; c += 256) { float e = expf(rin[c] - m); rout[c] = e; s += e; }
    s = blockSum256(s, sm);

    const float inv = 1.0f / s;
    float q = 0.f;
    for (int c = t; c < N; c += 256) { float v = rout[c] * inv; rout[c] = v; q += v * v; }
    q = blockSum256(q, sm);
    if (t == 0) sq[row] = q;
}

#if USE_TDM
// Issue a TDM 2D tile load: 128 rows x 32 dwords from a 4096x4096 f32 row-major
// tensor into LDS at lds_byte_addr, padding +4 dwords after every 32 dwords
// (=> LDS row stride 36 dwords == LDP). Descriptor per CDNA5 ISA section 8.
__device__ __forceinline__ void tdm_load_panel(unsigned lds_byte_addr, const float* gptr) {
    const unsigned long long ga = (unsigned long long)(uintptr_t)gptr;
    u32x4 g0;
    g0.x = 1u;                                            // count=1, user mode
    g0.y = lds_byte_addr;                                 // lds_addr
    g0.z = (unsigned)(ga & 0xFFFFFFFFu);                  // global_addr[31:0]
    g0.w = (unsigned)((ga >> 32) & 0x01FFFFFFu) | (2u << 30); // addr[56:32] | type=2
    i32x8 g1;
    g1[0] = (2 << 16)      // data_size = 4 bytes
          | (1 << 20)      // pad_enable
          | (4 << 22)      // pad_interval: every 32 dwords
          | (3 << 25);     // pad_amount: 4 dwords
    g1[1] = (int)(((unsigned)N & 0xFFFFu) << 16);         // tensor_dim0[15:0]
    g1[2] = (int)((((unsigned)N >> 16) & 0xFFFFu) | (((unsigned)N & 0xFFFFu) << 16));
    g1[3] = (int)((((unsigned)N >> 16) & 0xFFFFu) | (32u << 16)); // tile_dim0=32
    g1[4] = 128;                                          // tile_dim1=128, tile_dim2=0
    g1[5] = N;                                            // tensor_dim0_stride[31:0]
    g1[6] = 0;                                            // stride hi, dim1_stride lo
    g1[7] = 0;
    const i32x4 z4 = {0, 0, 0, 0};                        // D# groups 2/3 (2D tile)
    const i32x8 z8 = {0, 0, 0, 0, 0, 0, 0, 0};            // reserved arg (clang-23 form)
    __builtin_amdgcn_tensor_load_to_lds(g0, g1, z4, z4, z8, 0);
}
#endif

// C[i,j] = x2[i] + y2[j] - 2 * sum_k Xs[i,k]*Ys[j,k]   via V_WMMA_F32_16X16X4_F32.
// 256 threads = 8 waves (4x2), macro tile 128x128, 32x64 per wave (2x4 WMMA tiles).
// Double-buffered LDS; panels fed by the Tensor Data Mover (wave 0 issues).
__global__ void __launch_bounds__(256)
gemm_c_kernel(const float* __restrict__ Xs, const float* __restrict__ Ys,
              const float* __restrict__ x2, const float* __restrict__ y2,
              float* __restrict__ Cout) {
    __shared__ float smem[4 * PANEL_F];   // [A0][B0][A1][B1]

    const int t    = threadIdx.x;
    const int lane = t & 31;
    const int w    = t >> 5;
    const int wm   = w & 3;               // 0..3  (M direction, 32 rows each)
    const int wn   = w >> 2;              // 0..1  (N direction, 64 cols each)
    const int bm0  = blockIdx.y * BM;
    const int bn0  = blockIdx.x * BN;

    const int lm = lane & 15;             // row within 16x16 tile (A) / col (B)
    const int kh = (lane >> 4) << 1;      // K sub-offset per lane half

    v8f acc[2][4];
#pragma unroll
    for (int i = 0; i < 2; ++i)
#pragma unroll
        for (int j = 0; j < 4; ++j) acc[i][j] = (v8f){};

#if USE_TDM
    const unsigned smemBase = (unsigned)(uintptr_t)(void*)smem;
    // Prologue: wave 0 DMAs first K-block into buffer 0.
    if (w == 0) {
        tdm_load_panel(smemBase,                     Xs + (size_t)bm0 * N);
        tdm_load_panel(smemBase + PANEL_F * 4u,      Ys + (size_t)bn0 * N);
        WAIT_TENSOR0();
    }
    __syncthreads();
#else
    const int lrow = t >> 3;              // 0..31 staging row
    const int lc4  = (t & 7) << 2;        // 0..28 staging col (float4)
    {   // Prologue: synchronous stage of K-block 0 into buffer 0.
#pragma unroll
        for (int r = 0; r < 4; ++r) {
            const int row = lrow + r * 32;
            *(v4f*)&smem[row * LDP + lc4] =
                *(const v4f*)(Xs + (size_t)(bm0 + row) * N + lc4);
            *(v4f*)&smem[PANEL_F + row * LDP + lc4] =
                *(const v4f*)(Ys + (size_t)(bn0 + row) * N + lc4);
        }
    }
    __syncthreads();
#endif

    int p = 0;
    for (int k0 = 0; k0 < N; k0 += BK) {
        const float* As = smem + p * (2 * PANEL_F);
        const float* Bs = As + PANEL_F;
        const bool has_next = (k0 + BK) < N;

#if USE_TDM
        // Kick next K-block DMA into the other buffer; it overlaps the WMMAs below.
        if (w == 0 && has_next) {
            const unsigned nb = smemBase + (unsigned)((p ^ 1) * (2 * PANEL_F) * 4);
            tdm_load_panel(nb,                  Xs + (size_t)bm0 * N + (k0 + BK));
            tdm_load_panel(nb + PANEL_F * 4u,   Ys + (size_t)bn0 * N + (k0 + BK));
        }
#else
        v4f rA[4], rB[4];
        if (has_next) {
#pragma unroll
            for (int r = 0; r < 4; ++r) {
                const int row = lrow + r * 32;
                rA[r] = *(const v4f*)(Xs + (size_t)(bm0 + row) * N + (k0 + BK) + lc4);
                rB[r] = *(const v4f*)(Ys + (size_t)(bn0 + row) * N + (k0 + BK) + lc4);
            }
        }
#endif

#pragma unroll
        for (int kk = 0; kk < BK; kk += 4) {
            v2f a0 = *(const v2f*)&As[(wm * 32 +      lm) * LDP + kk + kh];
            v2f a1 = *(const v2f*)&As[(wm * 32 + 16 + lm) * LDP + kk + kh];
            v2f b0 = *(const v2f*)&Bs[(wn * 64 +      lm) * LDP + kk + kh];
            v2f b1 = *(const v2f*)&Bs[(wn * 64 + 16 + lm) * LDP + kk + kh];
            v2f b2 = *(const v2f*)&Bs[(wn * 64 + 32 + lm) * LDP + kk + kh];
            v2f b3 = *(const v2f*)&Bs[(wn * 64 + 48 + lm) * LDP + kk + kh];
            acc[0][0] = __builtin_amdgcn_wmma_f32_16x16x4_f32(false, a0, false, b0,
                                                              (short)0, acc[0][0], false, false);
            acc[0][1] = __builtin_amdgcn_wmma_f32_16x16x4_f32(false, a0, false, b1,
                                                              (short)0, acc[0][1], false, false);
            acc[0][2] = __builtin_amdgcn_wmma_f32_16x16x4_f32(false, a0, false, b2,
                                                              (short)0, acc[0][2], false, false);
            acc[0][3] = __builtin_amdgcn_wmma_f32_16x16x4_f32(false, a0, false, b3,
                                                              (short)0, acc[0][3], false, false);
            acc[1][0] = __builtin_amdgcn_wmma_f32_16x16x4_f32(false, a1, false, b0,
                                                              (short)0, acc[1][0], false, false);
            acc[1][1] = __builtin_amdgcn_wmma_f32_16x16x4_f32(false, a1, false, b1,
                                                              (short)0, acc[1][1], false, false);
            acc[1][2] = __builtin_amdgcn_wmma_f32_16x16x4_f32(false, a1, false, b2,
                                                              (short)0, acc[1][2], false, false);
            acc[1][3] = __builtin_amdgcn_wmma_f32_16x16x4_f32(false, a1, false, b3,
                                                              (short)0, acc[1][3], false, false);
        }

#if USE_TDM
        if (w == 0 && has_next) WAIT_TENSOR0();
#else
        if (has_next) {
            float* An = smem + (p ^ 1) * (2 * PANEL_F);
            float* Bn = An + PANEL_F;
#pragma unroll
            for (int r = 0; r < 4; ++r) {
                const int row = lrow + r * 32;
                *(v4f*)&An[row * LDP + lc4] = rA[r];
                *(v4f*)&Bn[row * LDP + lc4] = rB[r];
            }
        }
#endif
        __syncthreads();
        p ^= 1;
    }

    // Epilogue. C/D layout: VGPR r holds M = r + 8*(lane/16), N = lane%16.
    const int mbase = bm0 + wm * 32 + ((lane >> 4) << 3);
    const int nbase = bn0 + wn * 64 + lm;
#pragma unroll
    for (int ti = 0; ti < 2; ++ti) {
#pragma unroll
        for (int tj = 0; tj < 4; ++tj) {
            const int col = nbase + tj * 16;
            const float yv = y2[col];
#pragma unroll
            for (int r = 0; r < 8; ++r) {
                const int rowg = mbase + ti * 16 + r;
                Cout[(size_t)rowg * N + col] = x2[rowg] + yv - 2.0f * acc[ti][tj][r];
            }
        }
    }
}

// Streaming per-segment column logsumexp of (-C/EPS). Grid (N/256, SEG).
__global__ void __launch_bounds__(256)
col_lse_partial_kernel(const float* __restrict__ Cmat,
                       float* __restrict__ pm, float* __restrict__ ps) {
    const int col = blockIdx.x * 256 + threadIdx.x;
    const int seg = blockIdx.y;
    const int i0 = seg * (N / SEG);
    float m = -__builtin_inff(), s = 0.f;
    for (int i = i0; i < i0 + N / SEG; ++i) {
        const float v = -Cmat[(size_t)i * N + col] * INV_EPS;
        const float nm = fmaxf(m, v);
        s = s * expf(m - nm) + expf(v - nm);
        m = nm;
    }
    pm[(size_t)seg * N + col] = m;
    ps[(size_t)seg * N + col] = s;
}

__global__ void __launch_bounds__(256)
col_lse_combine_kernel(const float* __restrict__ pm, const float* __restrict__ ps,
                       float* __restrict__ Lc) {
    const int col = blockIdx.x * 256 + threadIdx.x;
    float m = -__builtin_inff();
    for (int sg = 0; sg < SEG; ++sg) m = fmaxf(m, pm[(size_t)sg * N + col]);
    float s = 0.f;
    for (int sg = 0; sg < SEG; ++sg)
        s += ps[(size_t)sg * N + col] * expf(pm[(size_t)sg * N + col] - m);
    Lc[col] = logf(s) + m;
}

// pi[i,j] = exp(LOG_NU - Lc[j] - C[i,j]/EPS); per-row partial of sum(pi*C).
__global__ void __launch_bounds__(256)
pi_cost_kernel(const float* __restrict__ Cmat, const float* __restrict__ Lc,
               float* __restrict__ Pi, float* __restrict__ cpart) {
    __shared__ float sm[256];
    const int row = blockIdx.x, t = threadIdx.x;
    const float* rC = Cmat + (size_t)row * N;
    float* rP = Pi + (size_t)row * N;
    float local = 0.f;
    for (int c = t; c < N; c += 256) {
        const float Cv = rC[c];
        const float p = expf(LOG_NU - Lc[c] - Cv * INV_EPS);
        rP[c] = p;
        local += p * Cv;
    }
    local = blockSum256(local, sm);
    if (t == 0) cpart[row] = local;
}

__global__ void __launch_bounds__(256)
cost_final_kernel(const float* __restrict__ cpart, float* __restrict__ out) {
    __shared__ float sm[256];
    const int t = threadIdx.x;
    float s = 0.f;
    for (int i = t; i < N; i += 256) s += cpart[i];
    s = blockSum256(s, sm);
    if (t == 0) out[0] = s;
}

extern "C" void kernel_launch(void* const* d_in, const int* in_sizes, int n_in,
                              void* d_out, int out_size, void* d_ws, size_t ws_size,
                              hipStream_t stream) {
    (void)in_sizes; (void)n_in; (void)out_size; (void)ws_size;
    const float* x = (const float*)d_in[0];
    const float* y = (const float*)d_in[1];

    float* out   = (float*)d_out;
    float* outPi = out + 1;             // N*N
    float* outC  = out + 1 + NNE;       // N*N

    float* ws    = (float*)d_ws;
    float* Xs    = ws;                  // N*N
    float* Ys    = Xs + NNE;            // N*N
    float* x2    = Ys + NNE;            // N
    float* y2    = x2 + N;              // N
    float* Lc    = y2 + N;              // N
    float* pm    = Lc + N;              // SEG*N
    float* ps    = pm + (size_t)SEG * N;  // SEG*N
    float* cpart = ps + (size_t)SEG * N;  // N

    softmax_rows_kernel<<<N, 256, 0, stream>>>(x, Xs, x2);
    softmax_rows_kernel<<<N, 256, 0, stream>>>(y, Ys, y2);
    gemm_c_kernel<<<dim3(N / BN, N / BM), 256, 0, stream>>>(Xs, Ys, x2, y2, outC);
    col_lse_partial_kernel<<<dim3(N / 256, SEG), 256, 0, stream>>>(outC, pm, ps);
    col_lse_combine_kernel<<<N / 256, 256, 0, stream>>>(pm, ps, Lc);
    pi_cost_kernel<<<N, 256, 0, stream>>>(outC, Lc, outPi, cpart);
    cost_final_kernel<<<1, 256, 0, stream>>>(cpart, out);
}